// AggrSum_46299747451335
// MI455X (gfx1250) — compile-verified
//
#include <hip/hip_runtime.h>

// Segment-sum / scatter-add:  out[node] += H[edge]  for each edge.
// H: [num_edges, 32] f32, idx: [num_edges] int, out: [node_num, 32] f32.
//
// Memory-bound: 256 MB streamed reads + 64M f32 atomics into a 12.8 MB
// output that stays resident in the 192 MB L2. Floor ~11 us @ 23.3 TB/s.
// CDNA5 path: pipelined GLOBAL_LOAD_ASYNC_TO_LDS_B128 (ASYNCcnt) with NT
// temporal hint, drained with s_wait_asynccnt; accumulation via relaxed
// agent-scope f32 atomics (global_atomic_add_f32, resolved in L2).

typedef float v4f __attribute__((ext_vector_type(4)));
typedef int   v4i __attribute__((vector_size(16)));   // matches builtin proto

typedef __attribute__((address_space(1))) v4i gv4i;   // global int4
typedef __attribute__((address_space(3))) v4i lv4i;   // LDS int4

#if defined(__has_builtin)
#if __has_builtin(__builtin_amdgcn_global_load_async_to_lds_b128)
#define USE_ASYNC_LDS 1
#endif
#endif
#ifndef USE_ASYNC_LDS
#define USE_ASYNC_LDS 0
#endif

// GFX12+ CPol: TH bits [2:0]; TH_NT = 1 (single-use stream, don't pollute L2).
#define CPOL_NT 1

__global__ __launch_bounds__(256) void AggrSum_46299747451335_kernel(
        const float* __restrict__ H,
        const int*   __restrict__ idx,
        float*       __restrict__ out,
        unsigned num_edges) {
    const v4f* __restrict__ H4 = (const v4f*)H;

    const unsigned n4     = num_edges * 8u;  // 32 floats/edge = 8 float4 chunks
    const unsigned stride = gridDim.x * blockDim.x;
    unsigned cur          = blockIdx.x * blockDim.x + threadIdx.x;

#if USE_ASYNC_LDS
    // Depth-2 async pipeline: per-thread ping-pong 16B slots. The async
    // engine fills slot[buf^1] for iteration k+1 while we consume slot[buf].
    // Async loads complete in order, so s_wait_asynccnt 1 guarantees the
    // older transfer has landed. s_wait_dscnt 0 before each issue drains the
    // previous ds_load of the slot being overwritten (DS and ASYNC ops are
    // mutually unordered in hardware).
    __shared__ v4f stage[2][256];

    if (cur < n4) {
        __builtin_amdgcn_global_load_async_to_lds_b128(
            (gv4i*)(H4 + cur), (lv4i*)&stage[0][threadIdx.x], 0, CPOL_NT);
    }
    int buf = 0;
    while (cur < n4) {
        const unsigned nxt = cur + stride;

        asm volatile("s_wait_dscnt 0x0" ::: "memory");
        if (nxt < n4) {
            __builtin_amdgcn_global_load_async_to_lds_b128(
                (gv4i*)(H4 + nxt), (lv4i*)&stage[buf ^ 1][threadIdx.x], 0, CPOL_NT);
            asm volatile("s_wait_asynccnt 0x1" ::: "memory");
        } else {
            asm volatile("s_wait_asynccnt 0x0" ::: "memory");
        }

        const v4f h = stage[buf][threadIdx.x];

        const unsigned edge = cur >> 3;
        const unsigned sub  = cur & 7u;
        const unsigned node = (unsigned)idx[edge];
        float* dst = out + (node * 32u + sub * 4u);
        __hip_atomic_fetch_add(dst + 0, h.x, __ATOMIC_RELAXED, __HIP_MEMORY_SCOPE_AGENT);
        __hip_atomic_fetch_add(dst + 1, h.y, __ATOMIC_RELAXED, __HIP_MEMORY_SCOPE_AGENT);
        __hip_atomic_fetch_add(dst + 2, h.z, __ATOMIC_RELAXED, __HIP_MEMORY_SCOPE_AGENT);
        __hip_atomic_fetch_add(dst + 3, h.w, __ATOMIC_RELAXED, __HIP_MEMORY_SCOPE_AGENT);

        buf ^= 1;
        cur = nxt;
    }
#else
    for (; cur < n4; cur += stride) {
        const v4f h = __builtin_nontemporal_load(H4 + cur);
        const unsigned edge = cur >> 3;
        const unsigned sub  = cur & 7u;
        const unsigned node = (unsigned)idx[edge];
        float* dst = out + (node * 32u + sub * 4u);
        __hip_atomic_fetch_add(dst + 0, h.x, __ATOMIC_RELAXED, __HIP_MEMORY_SCOPE_AGENT);
        __hip_atomic_fetch_add(dst + 1, h.y, __ATOMIC_RELAXED, __HIP_MEMORY_SCOPE_AGENT);
        __hip_atomic_fetch_add(dst + 2, h.z, __ATOMIC_RELAXED, __HIP_MEMORY_SCOPE_AGENT);
        __hip_atomic_fetch_add(dst + 3, h.w, __ATOMIC_RELAXED, __HIP_MEMORY_SCOPE_AGENT);
    }
#endif
}

__global__ __launch_bounds__(256) void AggrSum_zero_kernel(v4f* __restrict__ out, unsigned n4) {
    const unsigned i = blockIdx.x * blockDim.x + threadIdx.x;
    if (i < n4) {
        v4f z = {0.0f, 0.0f, 0.0f, 0.0f};
        out[i] = z;
    }
}

extern "C" void kernel_launch(void* const* d_in, const int* in_sizes, int n_in,
                              void* d_out, int out_size, void* d_ws, size_t ws_size,
                              hipStream_t stream) {
    const float* H   = (const float*)d_in[0];
    const int*   idx = (const int*)d_in[1];
    float*       out = (float*)d_out;

    const unsigned num_edges = (unsigned)in_sizes[1];

    // Zero the output (harness poisons d_out; atomics need a clean base).
    // out_size = node_num * 32, divisible by 4.
    const unsigned zn4 = (unsigned)out_size / 4u;
    unsigned zb = (zn4 + 255u) / 256u;
    if (zb < 1u) zb = 1u;
    AggrSum_zero_kernel<<<zb, 256, 0, stream>>>((v4f*)out, zn4);

    // One thread per 16B chunk of H; wave32 -> 4 edges (512B) per wave, fully
    // coalesced. Grid-stride with a capped launch (8192 blocks = 2M threads,
    // ~8 pipelined iterations per thread).
    const unsigned n4 = num_edges * 8u;
    unsigned sb = (n4 + 255u) / 256u;
    if (sb > 8192u) sb = 8192u;
    if (sb < 1u) sb = 1u;
    AggrSum_46299747451335_kernel<<<sb, 256, 0, stream>>>(H, idx, out, num_edges);
}